// AtomCenteredTensorMomentDescriptor_51539607552083
// MI455X (gfx1250) — compile-verified
//
#include <hip/hip_runtime.h>
#include <math.h>

// ---------------------------------------------------------------------------
// Types
// ---------------------------------------------------------------------------
typedef __attribute__((ext_vector_type(2))) float v2f;
typedef __attribute__((ext_vector_type(8))) float v8f;

struct Term { unsigned pk; float c; };   // pk = gidx | i<<8 | j<<13

#define WAVES 4           // waves per block; each wave owns a 16-edge tile
#define NTERM_DENSE 2919  // dense CG entry count over all 27 paths
#define GSTRIDE 164       // padded per-edge G stride (floats); 160 slots + pad

// ---------------------------------------------------------------------------
// Compile-time tables
// ---------------------------------------------------------------------------
constexpr int cPATHS[27][3] = {
  {0,0,0},{0,1,1},{0,2,2},{0,3,3},
  {1,0,1},{1,1,0},{1,1,2},{1,2,1},{1,2,3},{1,3,2},{1,3,4},
  {2,0,2},{2,1,1},{2,1,3},{2,2,0},{2,2,2},{2,2,4},{2,3,1},{2,3,3},
  {3,0,3},{3,1,2},{3,1,4},{3,2,1},{3,2,3},{3,3,0},{3,3,2},{3,3,4}};
constexpr int cPOFF[28] = {0,1,10,35,84,93,102,147,192,297,402,591,616,661,
                           766,791,916,1141,1246,1491,1540,1645,1834,1939,
                           2184,2233,2478,2919};
constexpr int cOFFIN[4]  = {0,1,4,9};
constexpr int cOFFOUT[5] = {0,1,4,9,16};
// output row k -> degree l3
constexpr int cK2L3[25] = {0,1,1,1,2,2,2,2,2,3,3,3,3,3,3,3,4,4,4,4,4,4,4,4,4};
// padded per-k offset into G slots (rows padded to multiple of 4: 4/8/8/8/4)
constexpr int cKOFFP[25] = {0,4,12,20,28,36,44,52,60,68,76,84,92,100,108,116,
                            124,128,132,136,140,144,148,152,156};  // total 160
// per-l3 pair count (unpadded) and number of float4 groups
constexpr int cL3NP[5] = {4,6,7,6,4};
constexpr int cNG[5]   = {1,2,2,2,1};
// per-l3 pair ids p = l1*4+l2, padded to 8 (pads use P[0] against G==0)
constexpr int cL3P8[5][8] = {
  {0,5,10,15,0,0,0,0},
  {1,4,6,9,11,14,0,0},
  {2,5,7,8,10,13,15,0},
  {3,6,9,11,12,14,0,0},
  {7,10,13,15,0,0,0,0}};

// ---------------------------------------------------------------------------
// Init: real-basis Clebsch-Gordan coefficients (Racah formula, f64)
// ---------------------------------------------------------------------------
__device__ static double factd(int n) {
  double f = 1.0;
  for (int i = 2; i <= n; ++i) f *= (double)i;
  return f;
}

__device__ static double cg_complex(int j1, int j2, int j3, int m1, int m2) {
  int m3 = m1 + m2;
  if (abs(m1) > j1 || abs(m2) > j2 || abs(m3) > j3) return 0.0;
  double pref = sqrt((double)(2 * j3 + 1) * factd(j1 + j2 - j3) *
                     factd(j1 - j2 + j3) * factd(-j1 + j2 + j3) /
                     factd(j1 + j2 + j3 + 1));
  pref *= sqrt(factd(j1 + m1) * factd(j1 - m1) * factd(j2 + m2) *
               factd(j2 - m2) * factd(j3 + m3) * factd(j3 - m3));
  int kmin = max(0, max(j2 - j3 - m1, j1 - j3 + m2));
  int kmax = min(j1 + j2 - j3, min(j1 - m1, j2 + m2));
  double s = 0.0;
  for (int k = kmin; k <= kmax; ++k) {
    double d = factd(k) * factd(j1 + j2 - j3 - k) * factd(j1 - m1 - k) *
               factd(j2 + m2 - k) * factd(j3 - j2 + m1 + k) *
               factd(j3 - j1 - m2 + k);
    s += ((k & 1) ? -1.0 : 1.0) / d;
  }
  return pref * s;
}

// nonzeros of row a of the complex->real SH unitary U(l); returns count (1|2)
__device__ static int urow(int l, int a, int* col, double* re, double* im) {
  const double is2 = 0.7071067811865475244;
  int m = a - l;
  if (m == 0) { col[0] = l; re[0] = 1.0; im[0] = 0.0; return 1; }
  if (m > 0) {
    double s = ((m & 1) ? -1.0 : 1.0) * is2;
    col[0] = l + m; re[0] = s;   im[0] = 0.0;
    col[1] = l - m; re[1] = is2; im[1] = 0.0;
    return 2;
  }
  int mm = -m;
  double s = ((mm & 1) ? -1.0 : 1.0) * is2;
  col[0] = l + mm; re[0] = 0.0; im[0] = -s;
  col[1] = l - mm; re[1] = 0.0; im[1] = is2;
  return 2;
}

__device__ static double real_cg(int l1, int l2, int l3, int a, int b, int c) {
  int c1[2], c2[2], c3[2];
  double r1[2], i1[2], r2[2], i2[2], r3[2], i3[2];
  int n1 = urow(l1, a, c1, r1, i1);
  int n2 = urow(l2, b, c2, r2, i2);
  int n3 = urow(l3, c, c3, r3, i3);
  double acc = 0.0;
  for (int x = 0; x < n1; ++x)
    for (int y = 0; y < n2; ++y)
      for (int z = 0; z < n3; ++z) {
        int m1 = c1[x] - l1, m2 = c2[y] - l2, m3 = c3[z] - l3;
        if (m1 + m2 != m3) continue;
        double cg = cg_complex(l1, l2, l3, m1, m2);
        if (cg == 0.0) continue;
        double pr = r1[x] * r2[y] - i1[x] * i2[y];
        double pi = r1[x] * i2[y] + i1[x] * r2[y];
        acc += (pr * r3[z] + pi * i3[z]) * cg;   // Re(u1*u2*conj(u3))
      }
  return acc;
}

// Parallel: one thread per dense CG entry
__global__ void cg_dense_kernel(Term* __restrict__ dense) {
  int t = blockIdx.x * blockDim.x + threadIdx.x;
  if (t >= NTERM_DENSE) return;
  int pi = 0;
  while (!(t >= cPOFF[pi] && t < cPOFF[pi + 1])) ++pi;
  int l1 = cPATHS[pi][0], l2 = cPATHS[pi][1], l3 = cPATHS[pi][2];
  int local = t - cPOFF[pi];
  int d3 = 2 * l3 + 1, d2 = 2 * l2 + 1;
  int c = local % d3;
  int b = (local / d3) % d2;
  int a = local / (d3 * d2);
  double v = real_cg(l1, l2, l3, a, b, c);
  int p = l1 * 4 + l2;
  int slot = 0;
  for (int s = 0; s < cL3NP[l3]; ++s) if (cL3P8[l3][s] == p) slot = s;
  int kg = cOFFOUT[l3] + c;
  unsigned pk = (unsigned)(cKOFFP[kg] + slot) |
                ((unsigned)(cOFFIN[l1] + a) << 8) |
                ((unsigned)(cOFFIN[l2] + b) << 13);
  dense[t].pk = pk;
  dense[t].c = (fabs(v) > 1e-10) ? (float)v : 0.0f;
}

// Serial compaction of nonzeros (cheap: 2919 iterations once per launch)
__global__ void cg_compact_kernel(const Term* __restrict__ dense,
                                  int* __restrict__ cnt,
                                  Term* __restrict__ terms, int maxT) {
  if (blockIdx.x != 0 || threadIdx.x != 0) return;
  int n = 0;
  for (int t = 0; t < NTERM_DENSE; ++t) {
    Term tm = dense[t];
    if (tm.c != 0.0f && n < maxT) { terms[n] = tm; ++n; }
  }
  *cnt = n;
}

// ---------------------------------------------------------------------------
// Accumulate + store, templated on bounds guarding so the common (full-tile)
// path has no divergent EXEC manipulation at all. Output stores are
// non-temporal: 419 MB write-once must not cycle through the 192 MB L2.
// ---------------------------------------------------------------------------
template <bool GUARD>
__device__ __forceinline__ void accum_store(const float* __restrict__ G,
                                            const v8f* t1c, const v8f* t2c,
                                            float* __restrict__ out,
                                            int e0, int ch, int lm, int hl,
                                            int E) {
#pragma unroll
  for (int v = 0; v < 8; ++v) {
    const int e = v + hl * 8;
    if (GUARD && (e0 + e >= E)) continue;
    float P[16];
#pragma unroll
    for (int l1 = 0; l1 < 4; ++l1)
#pragma unroll
      for (int l2 = 0; l2 < 4; ++l2)
        P[l1 * 4 + l2] = t1c[l1][v] * t2c[l2][v];
    const float* Ge = G + e * GSTRIDE;
    float* op = out + (size_t)(e0 + e) * 1600 + ch * 16 + lm;
#pragma unroll
    for (int k = 0; k < 25; ++k) {
      const int l3 = cK2L3[k];
      float acc = 0.0f;
#pragma unroll
      for (int g4 = 0; g4 < 2; ++g4) {
        if (g4 < cNG[l3]) {
          float4 gv = *(const float4*)(Ge + cKOFFP[k] + g4 * 4);
          acc += gv.x * P[cL3P8[l3][g4 * 4 + 0]];
          acc += gv.y * P[cL3P8[l3][g4 * 4 + 1]];
          acc += gv.z * P[cL3P8[l3][g4 * 4 + 2]];
          acc += gv.w * P[cL3P8[l3][g4 * 4 + 3]];
        }
      }
      __builtin_nontemporal_store(acc, op + (size_t)k * 64);
    }
  }
}

// ---------------------------------------------------------------------------
// Main kernel: 16-edge tile per wave; WMMA f32 16x16x4 for R @ [W1|W2]
// ---------------------------------------------------------------------------
__global__ void __launch_bounds__(WAVES * 32)
moment_kernel(const float* __restrict__ disp, const float* __restrict__ W1,
              const float* __restrict__ W2, const int* __restrict__ cntp,
              const Term* __restrict__ terms, float* __restrict__ out, int E) {
  __shared__ __align__(16) float Gs[WAVES][GSTRIDE * 16]; // CG bilinears/edge
  __shared__ float Ys[WAVES][16][16];                     // SH values per edge
  __shared__ float Rs[WAVES][16][8];                      // radial*cutoff/edge

  const int wave = threadIdx.x >> 5;
  const int lane = threadIdx.x & 31;
  const int hl   = lane >> 4;     // half-wave: 0 | 1
  const int lm   = lane & 15;
  const int tile = blockIdx.x * WAVES + wave;
  const int e0   = tile * 16;

  float* G = Gs[wave];
  {
    float4* Gz = (float4*)G;
    const float4 z4 = make_float4(0.f, 0.f, 0.f, 0.f);
    for (int i = lane; i < GSTRIDE * 16 / 4; i += 32) Gz[i] = z4;
  }

  if (lane < 16) {
    int e  = e0 + lane;
    int es = (e < E) ? e : 0;
    float x = __builtin_nontemporal_load(disp + es * 3 + 0);
    float y = __builtin_nontemporal_load(disp + es * 3 + 1);
    float z = __builtin_nontemporal_load(disp + es * 3 + 2);
    float r = sqrtf(x * x + y * y + z * z + 1e-12f);
    float inv = 1.0f / r;
    float ux = x * inv, uy = y * inv, uz = z * inv;
    float x2 = ux * ux, y2 = uy * uy, z2 = uz * uz;
    float* Yw = Ys[wave][lane];
    Yw[0]  = 0.28209479177387814f;
    Yw[1]  = 0.4886025119029199f * uy;
    Yw[2]  = 0.4886025119029199f * uz;
    Yw[3]  = 0.4886025119029199f * ux;
    Yw[4]  = 1.0925484305920792f * ux * uy;
    Yw[5]  = 1.0925484305920792f * uy * uz;
    Yw[6]  = 0.31539156525252005f * (3.0f * z2 - 1.0f);
    Yw[7]  = 1.0925484305920792f * ux * uz;
    Yw[8]  = 0.5462742152960396f * (x2 - y2);
    Yw[9]  = 0.5900435899266435f * uy * (3.0f * x2 - y2);
    Yw[10] = 2.890611442640554f * ux * uy * uz;
    Yw[11] = 0.4570457994644658f * uy * (5.0f * z2 - 1.0f);
    Yw[12] = 0.3731763325901154f * uz * (5.0f * z2 - 3.0f);
    Yw[13] = 0.4570457994644658f * ux * (5.0f * z2 - 1.0f);
    Yw[14] = 1.445305721320277f * uz * (x2 - y2);
    Yw[15] = 0.5900435899266435f * ux * (x2 - 3.0f * y2);
    float s2  = (r * 0.2f) * (r * 0.2f);
    float cut = (s2 < 1.0f) ? __expf(1.0f - 1.0f / (1.0f - s2)) : 0.0f;
    float q   = 0.6283185307179586477f * r;   // pi*r/CUTOFF
#pragma unroll
    for (int nn = 1; nn <= 8; ++nn) {
      float aa = q * (float)nn;
      Rs[wave][lane][nn - 1] = cut * __sinf(aa) / aa;
    }
  }
  __syncthreads();

  // Build G[e][gidx] = sum coeff * Y_i * Y_j  (half-waves split the term list)
  {
    const int nT = *cntp;
    const float* Yw = Ys[wave][lm];
    float* Ge = G + lm * GSTRIDE;
    for (int t = hl; t < nT; t += 2) {
      unsigned pk = terms[t].pk;
      float c = terms[t].c;
      int g = pk & 255, i = (pk >> 8) & 31, j = (pk >> 13) & 31;
      atomicAdd(&Ge[g], c * Yw[i] * Yw[j]);
    }
  }
  __syncthreads();

  // A operand (shared by all 16 GEMMs): 16x4 f32 tiles of R, K split by halves
  v2f aLo, aHi;
  aLo.x = Rs[wave][lm][2 * hl + 0];
  aLo.y = Rs[wave][lm][2 * hl + 1];
  aHi.x = Rs[wave][lm][4 + 2 * hl + 0];
  aHi.y = Rs[wave][lm][4 + 2 * hl + 1];

  const bool fullTile = (e0 + 16 <= E);   // uniform across the wave

  for (int ch = 0; ch < 4; ++ch) {        // 4 chunks of 16 features
    const int fcol = ch * 16 + lm;
    v8f t1c[4], t2c[4];
#pragma unroll
    for (int l = 0; l < 4; ++l) {
      const float* Wb = W1 + l * 512 + fcol;   // W1[l][n][g], n-stride 64
      v2f bLo, bHi;
      bLo.x = Wb[(2 * hl + 0) * 64];
      bLo.y = Wb[(2 * hl + 1) * 64];
      bHi.x = Wb[(4 + 2 * hl) * 64];
      bHi.y = Wb[(5 + 2 * hl) * 64];
      v8f c = {};
      c = __builtin_amdgcn_wmma_f32_16x16x4_f32(false, aLo, false, bLo,
                                                (short)0, c, false, false);
      c = __builtin_amdgcn_wmma_f32_16x16x4_f32(false, aHi, false, bHi,
                                                (short)0, c, false, false);
      t1c[l] = c;
      const float* Wc = W2 + l * 512 + fcol;
      bLo.x = Wc[(2 * hl + 0) * 64];
      bLo.y = Wc[(2 * hl + 1) * 64];
      bHi.x = Wc[(4 + 2 * hl) * 64];
      bHi.y = Wc[(5 + 2 * hl) * 64];
      v8f d = {};
      d = __builtin_amdgcn_wmma_f32_16x16x4_f32(false, aLo, false, bLo,
                                                (short)0, d, false, false);
      d = __builtin_amdgcn_wmma_f32_16x16x4_f32(false, aHi, false, bHi,
                                                (short)0, d, false, false);
      t2c[l] = d;
    }

    if (fullTile)
      accum_store<false>(G, t1c, t2c, out, e0, ch, lm, hl, E);
    else
      accum_store<true>(G, t1c, t2c, out, e0, ch, lm, hl, E);
  }
}

// ---------------------------------------------------------------------------
// Launch
// ---------------------------------------------------------------------------
extern "C" void kernel_launch(void* const* d_in, const int* in_sizes, int n_in,
                              void* d_out, int out_size, void* d_ws,
                              size_t ws_size, hipStream_t stream) {
  (void)n_in; (void)out_size;
  const float* disp = (const float*)d_in[0];
  const float* W1   = (const float*)d_in[1];
  const float* W2   = (const float*)d_in[2];
  float* out        = (float*)d_out;

  // ws layout: [0..15] header (cnt), [16..) compact terms, [32768..) dense
  int*  cnt   = (int*)d_ws;
  Term* terms = (Term*)((char*)d_ws + 16);
  Term* dense = (Term*)((char*)d_ws + 32768);
  int maxT = (int)((32768 - 16) / sizeof(Term));
  if ((size_t)(32768 + NTERM_DENSE * sizeof(Term)) > ws_size) {
    dense = terms;   // not expected; keep it legal
    maxT  = 0;
  }

  cg_dense_kernel<<<(NTERM_DENSE + 127) / 128, 128, 0, stream>>>(dense);
  cg_compact_kernel<<<1, 1, 0, stream>>>(dense, cnt, terms, maxT);

  const int E = in_sizes[0] / 3;            // 65536 edges
  const int tiles = (E + 15) / 16;
  const int blocks = (tiles + WAVES - 1) / WAVES;
  moment_kernel<<<blocks, WAVES * 32, 0, stream>>>(disp, W1, W2, cnt, terms,
                                                   out, E);
}